// Mamba_85779086836146
// MI455X (gfx1250) — compile-verified
//
#include <hip/hip_runtime.h>
#include <hip/hip_bf16.h>
#include <math.h>

// Problem dims (fixed by the reference)
#define B_     8
#define L_     1024
#define DM_    256
#define ED_    512          // DM*EXP
#define NST    16           // state dim N
#define RR     16           // rank R
#define DCONV_ 16
#define ML     (B_ * L_)    // 8192 rows in all GEMMs

typedef __attribute__((ext_vector_type(2))) float v2f;
typedef __attribute__((ext_vector_type(8))) float v8f;

__device__ __forceinline__ float sigmoidf_(float x) {
    return 1.f / (1.f + __expf(-x));
}

// ---------------------------------------------------------------------------
// Generic f32 WMMA GEMM:  C[m, n] = sum_k A[m*lda + k] * W[n*ldb + k]
// (weights stored [out_dim, in_dim], exactly as the reference einsums use)
// Each wave computes a 16 x (16*NT) tile with V_WMMA_F32_16X16X4_F32.
//
// f32 fragment layouts (CDNA5 ISA 7.12.2):
//   A 16x4 : lane<16 holds (M=lane, K=k0..k0+1) in v0..v1, lane>=16 K=k0+2..3
//   B 4x16 : lane<16 holds (N=lane, K rows k0,k0+1), lane>=16 rows k0+2,k0+3
//   C/D    : VGPR i -> row M = i + 8*(lane/16), col N = lane%16
//
// Software-pipelined: fragments for step k+4 are fetched before the WMMAs of
// step k issue, so each v_wmma waits only on the *previous* round of loads
// (s_wait_loadcnt >0) instead of the one just issued (loadcnt 0 stall).
// Requires: M%16==0, N%(16*NT)==0, K%4==0 and K>=4, rows 8-byte aligned.
// ---------------------------------------------------------------------------
template <int NT>
__global__ void gemm_wmma_f32(const float* __restrict__ A, int lda,
                              const float* __restrict__ W, int ldb,
                              float* __restrict__ C, int ldc,
                              int M, int N, int K)
{
    const int lane    = threadIdx.x & 31;
    const int wv      = (blockIdx.x * blockDim.x + threadIdx.x) >> 5;
    const int mtiles  = M >> 4;
    const int ngroups = N / (16 * NT);
    if (wv >= mtiles * ngroups) return;           // wave-uniform exit (EXEC stays full)

    const int mt = wv % mtiles;
    const int ng = wv / mtiles;
    const int m0 = mt << 4;
    const int n0 = ng * (16 * NT);
    const int hf = lane >> 4;                     // half-wave selector
    const int r  = lane & 15;

    const float* aRow = A + (size_t)(m0 + r) * lda + 2 * hf;
    const float* bRow[NT];
#pragma unroll
    for (int t = 0; t < NT; ++t)
        bRow[t] = W + (size_t)(n0 + t * 16 + r) * ldb + 2 * hf;

    v8f acc[NT];
    const v8f vzero = {0.f, 0.f, 0.f, 0.f, 0.f, 0.f, 0.f, 0.f};
#pragma unroll
    for (int t = 0; t < NT; ++t) acc[t] = vzero;

    // ---- prologue: fragments for k = 0
    v2f aCur = *(const v2f*)(aRow);
    v2f bCur[NT];
#pragma unroll
    for (int t = 0; t < NT; ++t) bCur[t] = *(const v2f*)(bRow[t]);

    // ---- steady state: fetch k+4 fragments, then multiply k fragments
    for (int k0 = 4; k0 < K; k0 += 4) {
        v2f aNxt = *(const v2f*)(aRow + k0);
        v2f bNxt[NT];
#pragma unroll
        for (int t = 0; t < NT; ++t) bNxt[t] = *(const v2f*)(bRow[t] + k0);

#pragma unroll
        for (int t = 0; t < NT; ++t)
            acc[t] = __builtin_amdgcn_wmma_f32_16x16x4_f32(
                false, aCur, false, bCur[t], (short)0, acc[t], false, false);

        aCur = aNxt;
#pragma unroll
        for (int t = 0; t < NT; ++t) bCur[t] = bNxt[t];
    }

    // ---- epilogue: last k-step
#pragma unroll
    for (int t = 0; t < NT; ++t)
        acc[t] = __builtin_amdgcn_wmma_f32_16x16x4_f32(
            false, aCur, false, bCur[t], (short)0, acc[t], false, false);

#pragma unroll
    for (int t = 0; t < NT; ++t)
#pragma unroll
        for (int i = 0; i < 8; ++i)
            C[(size_t)(m0 + i + 8 * hf) * ldc + (n0 + t * 16 + r)] = acc[t][i];
}

// ---------------------------------------------------------------------------
// Depthwise causal conv (DCONV taps) + bias + SiLU.
// Block = one (b,l) row, thread = channel e (512 threads, coalesced over e).
// xf lives in the low half of xz rows (row stride 2*ED).
// ---------------------------------------------------------------------------
__global__ void conv_silu_kernel(const float* __restrict__ xz,
                                 const float* __restrict__ Wc,   // [ED, DCONV]
                                 const float* __restrict__ bc,   // [ED]
                                 float* __restrict__ xf_act)     // [ML, ED]
{
    const int e  = threadIdx.x;          // 0..511
    const int bl = blockIdx.x;           // b*L + l
    const int l  = bl & (L_ - 1);

    float w[DCONV_];
#pragma unroll
    for (int t = 0; t < DCONV_; ++t) w[t] = Wc[e * DCONV_ + t];

    float acc = bc[e];
#pragma unroll
    for (int t = 0; t < DCONV_; ++t) {
        const int lt = l - (DCONV_ - 1) + t;     // causal left pad
        if (lt >= 0)
            acc += xz[(size_t)(bl - (DCONV_ - 1) + t) * (2 * ED_) + e] * w[t];
    }
    acc = acc * sigmoidf_(acc);                  // SiLU
    xf_act[(size_t)bl * ED_ + e] = acc;
}

// delta = softplus(delta_pre + b_dt[e]), in place over [ML, ED]
__global__ void softplus_bias_kernel(float* __restrict__ delta,
                                     const float* __restrict__ b_dt)
{
    const int idx = blockIdx.x * blockDim.x + threadIdx.x;
    const int e   = idx & (ED_ - 1);
    const float p = delta[idx] + b_dt[e];
    delta[idx]    = (p > 20.f) ? p : log1pf(__expf(p));
}

// ---------------------------------------------------------------------------
// Selective scan, parallel over (b, e, n): lane owns one state h[e][n].
// Per step: h = exp(delta*A[e,n])*h + delta*B[n]*xf ; y = sum_n h*C[n]
// (butterfly __shfl_xor over the 16-lane n-group -> ds permute network).
// Fused epilogue: g = (y + D[e]*xf) * silu(zf)  -- y never touches HBM.
// __builtin_prefetch -> global_prefetch_b8 pulls rows a few steps ahead.
// ---------------------------------------------------------------------------
#define SCAN_PF 8

__global__ void scan_kernel(const float* __restrict__ delta,   // [ML, ED]
                            const float* __restrict__ xf,      // [ML, ED]
                            const float* __restrict__ dBC,     // [ML, 48]
                            const float* __restrict__ xz,      // [ML, 2*ED] (zf half)
                            const float* __restrict__ A_log,   // [ED, NST]
                            const float* __restrict__ Dp,      // [ED]
                            float* __restrict__ g)             // [ML, ED]
{
    const int tid = blockIdx.x * blockDim.x + threadIdx.x;     // B*ED*NST threads
    const int n   = tid & (NST - 1);
    const int e   = (tid >> 4) & (ED_ - 1);
    const int b   = tid >> 13;                                 // 4 (n) + 9 (e)

    const float Aen = -__expf(A_log[e * NST + n]);
    const float De  = Dp[e];
    float h = 0.f;

    const size_t base = (size_t)b * L_;
    for (int l = 0; l < L_; ++l) {
        const size_t bl = base + l;

        if (l + SCAN_PF < L_) {                  // speculative prefetch ahead
            const size_t blp = bl + SCAN_PF;
            __builtin_prefetch(&dBC[blp * 48 + RR + n], 0, 0);
            if (n == 0) {
                __builtin_prefetch(&delta[blp * ED_ + e], 0, 0);
                __builtin_prefetch(&xf[blp * ED_ + e], 0, 0);
            }
        }

        const float dlt = delta[bl * ED_ + e];         // uniform over n-group
        const float xv  = xf[bl * ED_ + e];
        const float Bn  = dBC[bl * 48 + RR + n];
        const float Cn  = dBC[bl * 48 + RR + NST + n];

        const float dA = __expf(dlt * Aen);
        h = dA * h + dlt * Bn * xv;
        float yn = h * Cn;

        // reduce over the 16 states (stays within wave32 halves: masks < 16)
        yn += __shfl_xor(yn, 1);
        yn += __shfl_xor(yn, 2);
        yn += __shfl_xor(yn, 4);
        yn += __shfl_xor(yn, 8);

        if (n == 0) {
            const float z = xz[bl * (2 * ED_) + ED_ + e];      // zf
            g[bl * ED_ + e] = (yn + De * xv) * (z * sigmoidf_(z));
        }
    }
}

// ---------------------------------------------------------------------------
extern "C" void kernel_launch(void* const* d_in, const int* in_sizes, int n_in,
                              void* d_out, int out_size, void* d_ws, size_t ws_size,
                              hipStream_t stream)
{
    (void)in_sizes; (void)n_in; (void)out_size; (void)ws_size;

    const float* x      = (const float*)d_in[0];   // [B,L,DM]
    const float* W_in   = (const float*)d_in[1];   // [2*ED, DM]
    const float* W_conv = (const float*)d_in[2];   // [ED, DCONV]
    const float* b_conv = (const float*)d_in[3];   // [ED]
    const float* W_x    = (const float*)d_in[4];   // [R+2N, ED]
    const float* W_dt   = (const float*)d_in[5];   // [ED, R]
    const float* b_dt   = (const float*)d_in[6];   // [ED]
    const float* A_log  = (const float*)d_in[7];   // [ED, N]
    const float* Dv     = (const float*)d_in[8];   // [ED]
    const float* W_out  = (const float*)d_in[9];   // [DM, ED]
    float* out = (float*)d_out;                    // [B,L,DM] f32

    // workspace layout (floats), all chunks 16-float aligned
    float* ws     = (float*)d_ws;
    float* xz     = ws;                                // ML * 1024  (32 MB)
    float* xf_act = xz     + (size_t)ML * (2 * ED_);   // ML * 512   (16 MB)
    float* dBC    = xf_act + (size_t)ML * ED_;         // ML * 48    (1.5 MB)
    float* delta  = dBC    + (size_t)ML * 48;          // ML * 512   (16 MB)
    float* g      = delta  + (size_t)ML * ED_;         // ML * 512   (16 MB)

    // 1) xz = x @ W_in^T          M=8192 N=1024 K=256
    {
        const int waves = (ML / 16) * ((2 * ED_) / 64);
        gemm_wmma_f32<4><<<dim3((waves * 32 + 127) / 128), dim3(128), 0, stream>>>(
            x, DM_, W_in, DM_, xz, 2 * ED_, ML, 2 * ED_, DM_);
    }
    // 2) xf = silu(depthwise_conv(xf) + b_conv)
    conv_silu_kernel<<<dim3(ML), dim3(ED_), 0, stream>>>(xz, W_conv, b_conv, xf_act);

    // 3) dBC = xf @ W_x^T         M=8192 N=48 K=512
    {
        const int waves = (ML / 16) * 1;
        gemm_wmma_f32<3><<<dim3((waves * 32 + 127) / 128), dim3(128), 0, stream>>>(
            xf_act, ED_, W_x, ED_, dBC, 48, ML, 48, ED_);
    }
    // 4) delta_pre = dBC[:, :R] @ W_dt^T   M=8192 N=512 K=16 (lda=48 slices R cols)
    {
        const int waves = (ML / 16) * (ED_ / 64);
        gemm_wmma_f32<4><<<dim3((waves * 32 + 127) / 128), dim3(128), 0, stream>>>(
            dBC, 48, W_dt, RR, delta, ED_, ML, ED_, RR);
    }
    // 5) delta = softplus(delta_pre + b_dt)
    softplus_bias_kernel<<<dim3((ML * ED_) / 256), dim3(256), 0, stream>>>(delta, b_dt);

    // 6) scan + fused (y + D*xf) * silu(zf)
    scan_kernel<<<dim3((B_ * ED_ * NST) / 256), dim3(256), 0, stream>>>(
        delta, xf_act, dBC, xz, A_log, Dv, g);

    // 7) out = g @ W_out^T        M=8192 N=256 K=512
    {
        const int waves = (ML / 16) * (DM_ / 64);
        gemm_wmma_f32<4><<<dim3((waves * 32 + 127) / 128), dim3(128), 0, stream>>>(
            g, ED_, W_out, ED_, out, DM_, ML, DM_, ED_);
    }
}